// SimilarityEBR_6442450944543
// MI455X (gfx1250) — compile-verified
//
#include <hip/hip_runtime.h>
#include <hip/hip_bf16.h>

typedef __attribute__((ext_vector_type(2))) float v2f;
typedef __attribute__((ext_vector_type(8))) float v8f;

#define BM_BITS  (1u << 17)          // bitmap covers ids in [0, 131072)
#define BM_WORDS (BM_BITS / 32)      // 4096 u32 = 16 KB per bitmap
#define WAVES_PER_BLOCK 8
#define CAND_CAP 1024
#define SORT_CAP 2048

__device__ __forceinline__ bool bm_test(const unsigned* __restrict__ bm, int v) {
    unsigned u = (unsigned)v;
    if (u >= BM_BITS) return false;
    return (bm[u >> 5] >> (u & 31)) & 1u;
}

__device__ __forceinline__ unsigned float_to_key(float f) {
    unsigned u = __float_as_uint(f);
    return (u & 0x80000000u) ? ~u : (u | 0x80000000u);
}

__device__ __forceinline__ float key_to_float(unsigned k) {
    unsigned u = (k & 0x80000000u) ? (k ^ 0x80000000u) : ~k;
    return __uint_as_float(u);
}

// ---------------- init: zero bitmaps / hist / counters, seed select state ----
__global__ void ebr_init_kernel(unsigned* bm1, unsigned* bm2, unsigned* bmL,
                                unsigned* hist, unsigned* sel, unsigned* cnts,
                                const int* __restrict__ topkp) {
    int i = blockIdx.x * blockDim.x + threadIdx.x;
    if (i < (int)BM_WORDS) { bm1[i] = 0u; bm2[i] = 0u; bmL[i] = 0u; }
    if (i < 256) hist[i] = 0u;
    if (i == 0) {
        sel[0] = 0u;                       // prefix
        sel[1] = (unsigned)(*topkp);       // remaining k
        cnts[0] = 0u; cnts[1] = 0u;        // gt / eq counters
    }
}

// ---------------- scatter query-filter ids into bitmaps -----------------------
__global__ void ebr_scatter_kernel(const int* __restrict__ qf1, int n1,
                                   const int* __restrict__ qf2, int n2,
                                   const int* __restrict__ qf3, int n3,
                                   unsigned* bm1, unsigned* bm2, unsigned* bmL) {
    int i = blockIdx.x * blockDim.x + threadIdx.x;
    int total = n1 + n2 + n3;
    if (i >= total) return;
    int v; unsigned* bm;
    if (i < n1)           { v = qf1[i];           bm = bm1; }
    else if (i < n1 + n2) { v = qf2[i - n1];      bm = bm2; }
    else                  { v = qf3[i - n1 - n2]; bm = bmL; }
    unsigned u = (unsigned)v;
    if (u < BM_BITS) atomicOr(&bm[u >> 5], 1u << (u & 31));
}

// ---------------- main pass: WMMA dot products + filter -> sortable keys -----
__global__ __launch_bounds__(256)
void ebr_score_kernel(const float* __restrict__ emb, const float* __restrict__ q,
                      const int* __restrict__ attrs, const int* __restrict__ qf0p,
                      const unsigned* __restrict__ bm1, const unsigned* __restrict__ bm2,
                      const unsigned* __restrict__ bmL, unsigned* __restrict__ keys,
                      int N) {
    __shared__ float s_q[128];
    __shared__ float s_sc[WAVES_PER_BLOCK][32];

    int tid = threadIdx.x;
    if (tid < 128) s_q[tid] = q[tid];
    __syncthreads();

    int lane = tid & 31;
    int wv   = tid >> 5;
    int tile = blockIdx.x * WAVES_PER_BLOCK + wv;   // 32 items per wave-tile
    int full_tiles = N >> 5;
    int qf0 = *qf0p;

    float sc = 0.0f;
    int item = tile * 32 + lane;

    if (tile < full_tiles) {
        // V_WMMA_F32_16X16X4_F32: A 16x4 f32 (2 VGPRs), B 4x16 f32 (2 VGPRs).
        // Per ISA 7.12.2: lanes 0-15 hold rows M=0..15 with K={k,k+1},
        // lanes 16-31 hold the same rows with K={k+2,k+3}.
        int m  = lane & 15;
        int hi = lane >> 4;
        size_t row0 = (size_t)(tile * 32 + m) * 128 + (size_t)(hi * 2);
        size_t row1 = row0 + (size_t)16 * 128;
        const float* qh = &s_q[hi * 2];

        v8f c0 = {0.f,0.f,0.f,0.f,0.f,0.f,0.f,0.f};
        v8f c1 = {0.f,0.f,0.f,0.f,0.f,0.f,0.f,0.f};

        #pragma unroll 8
        for (int k = 0; k < 128; k += 4) {
            v2f a0 = *reinterpret_cast<const v2f*>(emb + row0 + k);
            v2f a1 = *reinterpret_cast<const v2f*>(emb + row1 + k);
            v2f b;  // query replicated to all 16 B columns -> every D column = dots
            b.x = qh[k];
            b.y = qh[k + 1];
            c0 = __builtin_amdgcn_wmma_f32_16x16x4_f32(false, a0, false, b,
                                                       (short)0, c0, false, false);
            c1 = __builtin_amdgcn_wmma_f32_16x16x4_f32(false, a1, false, b,
                                                       (short)0, c1, false, false);
        }

        // D layout: lane group (0-15 vs 16-31) holds M=j / M=j+8 in VGPR j.
        if (m == 0) {   // lanes 0 and 16 publish one column of D
            #pragma unroll
            for (int j = 0; j < 8; ++j) {
                s_sc[wv][hi * 8 + j]      = c0[j];
                s_sc[wv][16 + hi * 8 + j] = c1[j];
            }
        }
        __builtin_amdgcn_wave_barrier();
        sc = s_sc[wv][lane];
    } else {
        if (item >= N) return;
        const float* r = emb + (size_t)item * 128;
        float acc = 0.f;
        #pragma unroll 8
        for (int k = 0; k < 128; ++k) acc += r[k] * s_q[k];
        sc = acc;
    }

    if (item < N) {
        const int* ap = attrs + (size_t)item * 5;
        int a0 = ap[0], a1 = ap[1], a2 = ap[2], a3 = ap[3], a4 = ap[4];
        bool ok = (a0 > qf0) && (a1 > 0) &&
                  (bm_test(bm1, a2) || bm_test(bm2, a3)) &&
                  bm_test(bmL, a4);
        unsigned key = ok ? float_to_key(sc) : 0x007FFFFFu;  // key(-inf)
        keys[item] = key;
    }
}

// ---------------- radix-select: per-byte histogram -----------------------------
__global__ __launch_bounds__(256)
void ebr_hist_kernel(const unsigned* __restrict__ keys, int N,
                     const unsigned* __restrict__ sel, unsigned* hist, int byte) {
    __shared__ unsigned h[256];
    h[threadIdx.x] = 0u;
    __syncthreads();
    unsigned mask = (byte == 3) ? 0u : (0xFFFFFFFFu << ((byte + 1) * 8));
    unsigned pref = sel[0] & mask;
    int shift = byte * 8;
    for (int i = blockIdx.x * blockDim.x + threadIdx.x; i < N;
         i += gridDim.x * blockDim.x) {
        unsigned k = keys[i];
        if ((k & mask) == pref) atomicAdd(&h[(k >> shift) & 255u], 1u);
    }
    __syncthreads();
    unsigned c = h[threadIdx.x];
    if (c) atomicAdd(&hist[threadIdx.x], c);
}

// ---------------- radix-select: pick digit, update prefix/remK -----------------
__global__ __launch_bounds__(256)
void ebr_select_kernel(unsigned* hist, unsigned* sel, int byte) {
    if (threadIdx.x == 0) {
        unsigned rem = sel[1];
        unsigned acc = 0u;
        int chosen = 0;
        for (int d = 255; d >= 0; --d) {
            unsigned c = hist[d];
            if (acc + c >= rem) { chosen = d; break; }
            acc += c;
        }
        sel[0] |= (unsigned)chosen << (byte * 8);
        sel[1] = rem - acc;
    }
    __syncthreads();
    hist[threadIdx.x] = 0u;   // ready for next pass
}

// ---------------- gather candidates around the threshold -----------------------
__global__ __launch_bounds__(256)
void ebr_compact_kernel(const unsigned* __restrict__ keys, int N,
                        const unsigned* __restrict__ sel, unsigned* cnts,
                        unsigned long long* candGt, unsigned long long* candEq) {
    unsigned T = sel[0];
    for (int i = blockIdx.x * blockDim.x + threadIdx.x; i < N;
         i += gridDim.x * blockDim.x) {
        unsigned k = keys[i];
        if (k > T) {
            unsigned p = atomicAdd(&cnts[0], 1u);
            if (p < CAND_CAP)
                candGt[p] = ((unsigned long long)k << 32) | (unsigned)(~(unsigned)i);
        } else if (k == T) {
            unsigned p = atomicAdd(&cnts[1], 1u);
            if (p < CAND_CAP)
                candEq[p] = ((unsigned long long)k << 32) | (unsigned)(~(unsigned)i);
        }
    }
}

// ---------------- single-block bitonic sort (desc key, asc index) + output ----
__global__ __launch_bounds__(256)
void ebr_sort_out_kernel(const unsigned long long* __restrict__ candGt,
                         const unsigned long long* __restrict__ candEq,
                         const unsigned* __restrict__ cnts,
                         const int* __restrict__ topkp,
                         const int* __restrict__ item_ids, int N,
                         float* __restrict__ out, int out_size) {
    __shared__ unsigned long long arr[SORT_CAP];

    int k = *topkp;
    int k2 = out_size / 2;
    if (k2 < k) k = k2;
    if (k > CAND_CAP) k = CAND_CAP;

    unsigned ngt = cnts[0]; if (ngt > CAND_CAP) ngt = CAND_CAP;
    unsigned neq = cnts[1]; if (neq > CAND_CAP) neq = CAND_CAP;

    for (int i = threadIdx.x; i < SORT_CAP; i += blockDim.x) {
        unsigned long long v = 0ull;
        if (i < (int)ngt)              v = candGt[i];
        else if (i < (int)(ngt + neq)) v = candEq[i - ngt];
        arr[i] = v;
    }
    __syncthreads();

    // bitonic sort, descending on ((key<<32) | ~idx): key desc, tie -> idx asc
    for (int ks = 2; ks <= SORT_CAP; ks <<= 1) {
        for (int j = ks >> 1; j > 0; j >>= 1) {
            for (int i = threadIdx.x; i < SORT_CAP; i += blockDim.x) {
                int ixj = i ^ j;
                if (ixj > i) {
                    unsigned long long a = arr[i], b = arr[ixj];
                    bool desc = ((i & ks) == 0);
                    if (desc ? (a < b) : (a > b)) { arr[i] = b; arr[ixj] = a; }
                }
            }
            __syncthreads();
        }
    }

    for (int i = threadIdx.x; i < k; i += blockDim.x) {
        unsigned long long v = arr[i];
        unsigned key = (unsigned)(v >> 32);
        unsigned idx = ~(unsigned)v;
        out[i] = key_to_float(key);
        float idf = 0.f;
        if (idx < (unsigned)N) idf = (float)item_ids[idx];
        out[k + i] = idf;
    }
}

extern "C" void kernel_launch(void* const* d_in, const int* in_sizes, int n_in,
                              void* d_out, int out_size, void* d_ws, size_t ws_size,
                              hipStream_t stream) {
    const float* query = (const float*)d_in[0];
    const int*   qf0p  = (const int*)d_in[1];
    const int*   qf1   = (const int*)d_in[2];
    const int*   qf2   = (const int*)d_in[3];
    const int*   qf3   = (const int*)d_in[4];
    const int*   topkp = (const int*)d_in[5];
    const float* emb   = (const float*)d_in[6];
    const int*   attrs = (const int*)d_in[7];
    const int*   ids   = (const int*)d_in[8];

    int n1 = in_sizes[2], n2 = in_sizes[3], n3 = in_sizes[4];
    int N  = in_sizes[8];

    // workspace layout (256-B aligned slices)
    char* ws = (char*)d_ws;
    size_t off = 0;
    auto take = [&](size_t bytes) -> char* {
        char* p = ws + off;
        off += (bytes + 255) & ~(size_t)255;
        return p;
    };
    unsigned* keys = (unsigned*)take((size_t)N * 4);
    unsigned* bm1  = (unsigned*)take(BM_WORDS * 4);
    unsigned* bm2  = (unsigned*)take(BM_WORDS * 4);
    unsigned* bmL  = (unsigned*)take(BM_WORDS * 4);
    unsigned* hist = (unsigned*)take(256 * 4);
    unsigned* sel  = (unsigned*)take(64);
    unsigned* cnts = (unsigned*)take(64);
    unsigned long long* candGt = (unsigned long long*)take(CAND_CAP * 8);
    unsigned long long* candEq = (unsigned long long*)take(CAND_CAP * 8);
    (void)ws_size;

    ebr_init_kernel<<<(BM_WORDS + 255) / 256, 256, 0, stream>>>(
        bm1, bm2, bmL, hist, sel, cnts, topkp);

    int tot = n1 + n2 + n3;
    ebr_scatter_kernel<<<(tot + 255) / 256, 256, 0, stream>>>(
        qf1, n1, qf2, n2, qf3, n3, bm1, bm2, bmL);

    int tiles = (N + 31) / 32;
    ebr_score_kernel<<<(tiles + WAVES_PER_BLOCK - 1) / WAVES_PER_BLOCK, 256, 0, stream>>>(
        emb, query, attrs, qf0p, bm1, bm2, bmL, keys, N);

    for (int b = 3; b >= 0; --b) {
        ebr_hist_kernel<<<1024, 256, 0, stream>>>(keys, N, sel, hist, b);
        ebr_select_kernel<<<1, 256, 0, stream>>>(hist, sel, b);
    }

    ebr_compact_kernel<<<2048, 256, 0, stream>>>(keys, N, sel, cnts, candGt, candEq);

    ebr_sort_out_kernel<<<1, 256, 0, stream>>>(
        candGt, candEq, cnts, topkp, ids, N, (float*)d_out, out_size);
}